// CTM_21973052686906
// MI455X (gfx1250) — compile-verified
//
#include <hip/hip_runtime.h>
#include <math.h>

#define BB  16
#define NN  1024
#define CC  1024
#define KC  256   // cluster_num = N/4
#define KNN 5
#define FLT_BIG 3.402823466e38f
#define PAD 20    // 80B row stride: 16B-aligned for B128 async, bank-conflict-free

typedef __attribute__((ext_vector_type(2))) float v2f;
typedef __attribute__((ext_vector_type(8))) float v8f;

// ---------------- 1. row squared norms ----------------
__global__ void k_norms(const float* __restrict__ x, float* __restrict__ x2) {
    int row = blockIdx.x;              // b*N + n
    int tid = threadIdx.x;             // 256
    const float* p = x + (size_t)row * CC;
    float s = 0.f;
    for (int c = tid; c < CC; c += 256) { float v = p[c]; s += v * v; }
    __shared__ float red[256];
    red[tid] = s; __syncthreads();
    for (int off = 128; off > 0; off >>= 1) {
        if (tid < off) red[tid] += red[tid + off];
        __syncthreads();
    }
    if (tid == 0) x2[row] = red[0];
}

// ---------------- 2. WMMA fp32 Gram -> distance matrix ----------------
// block = 128 threads (4 waves), tile 64x64, K chunk = 16, double-buffered
// LDS staged with GLOBAL_LOAD_ASYNC_TO_LDS_B128 (ASYNCcnt) overlapping WMMA.
__device__ __forceinline__ void issue_chunk(const float* __restrict__ xb,
                                            int R, int Cb, int k0,
                                            float (*la)[PAD], float (*lb)[PAD],
                                            int tid) {
    // 2 tiles x 64 rows x 64B = 512 B128 transfers, 4 per thread
    #pragma unroll
    for (int it = 0; it < 4; ++it) {
        int i   = tid + it * 128;          // 0..511
        int mat = i >> 8;                  // 0 = A tile, 1 = B tile
        int r   = (i >> 2) & 63;
        int seg = i & 3;                   // 16B segment within the 64B row chunk
        const float* gp = xb + (size_t)((mat ? Cb : R) + r) * CC + k0 + seg * 4;
        float* lp = mat ? &lb[r][seg * 4] : &la[r][seg * 4];
        unsigned laddr = (unsigned)(uintptr_t)lp;                  // LDS byte offset
        unsigned long long gaddr = (unsigned long long)(uintptr_t)gp;
        asm volatile("global_load_async_to_lds_b128 %0, %1, off"
                     :: "v"(laddr), "v"(gaddr) : "memory");
    }
}

__global__ __launch_bounds__(128)
void k_dist(const float* __restrict__ x, const float* __restrict__ x2,
            float* __restrict__ dist) {
    const int b    = blockIdx.z;
    const int R    = blockIdx.y * 64;
    const int Cb   = blockIdx.x * 64;
    const int tid  = threadIdx.x;
    const int wave = tid >> 5;
    const int lane = tid & 31;
    const int laneM = lane & 15;
    const int koff  = (lane >> 4) << 1;    // lanes 0-15: K base 0 ; 16-31: K base 2

    __shared__ __align__(16) float la[2][64][PAD];
    __shared__ __align__(16) float lb[2][64][PAD];

    const float* xb = x + (size_t)b * NN * CC;

    v8f acc[4] = {};

    // prologue: fetch chunk 0
    issue_chunk(xb, R, Cb, 0, la[0], lb[0], tid);
    asm volatile("s_wait_asynccnt 0" ::: "memory");
    __syncthreads();

    int cur = 0;
    for (int k0 = 0; k0 < CC; k0 += 16) {
        int nxt = cur ^ 1;
        if (k0 + 16 < CC)                   // prefetch next chunk while computing
            issue_chunk(xb, R, Cb, k0 + 16, la[nxt], lb[nxt], tid);

        #pragma unroll
        for (int kk = 0; kk < 4; ++kk) {
            const int kb = kk * 4 + koff;
            v2f a;                           // A: 16x4 fp32 fragment
            a.x = la[cur][wave * 16 + laneM][kb];
            a.y = la[cur][wave * 16 + laneM][kb + 1];
            #pragma unroll
            for (int j = 0; j < 4; ++j) {
                v2f bf;                      // B: 4x16 fp32 fragment (x^T)
                bf.x = lb[cur][j * 16 + laneM][kb];
                bf.y = lb[cur][j * 16 + laneM][kb + 1];
                acc[j] = __builtin_amdgcn_wmma_f32_16x16x4_f32(
                    false, a, false, bf, (short)0, acc[j], false, false);
            }
        }

        asm volatile("s_wait_asynccnt 0" ::: "memory");  // next buffer landed
        __syncthreads();                                  // everyone done reading cur
        cur = nxt;
    }

    // epilogue: dist = sqrt(max(x2_i + x2_j - 2G, 0)) / sqrt(1024)
    const int hi = (lane >> 4) * 8;
    #pragma unroll
    for (int j = 0; j < 4; ++j) {
        #pragma unroll
        for (int t = 0; t < 8; ++t) {
            int gi = R + wave * 16 + t + hi;
            int gj = Cb + j * 16 + laneM;
            float d2 = x2[b * NN + gi] + x2[b * NN + gj] - 2.0f * acc[j][t];
            dist[((size_t)b * NN + gi) * NN + gj] = sqrtf(fmaxf(d2, 0.f)) * 0.03125f;
        }
    }
}

// ---------------- 3. kNN(5) density ----------------
__global__ void k_density(const float* __restrict__ dist, float* __restrict__ density) {
    int row = blockIdx.x;              // b*N + i
    int tid = threadIdx.x;             // 128
    const float* dr = dist + (size_t)row * NN;
    float best[KNN];
    #pragma unroll
    for (int t = 0; t < KNN; ++t) best[t] = FLT_BIG;
    for (int j = tid; j < NN; j += 128) {
        float d = dr[j];
        if (d < best[KNN - 1]) {
            best[KNN - 1] = d;
            #pragma unroll
            for (int t = KNN - 1; t > 0; --t)
                if (best[t] < best[t - 1]) { float tp = best[t]; best[t] = best[t - 1]; best[t - 1] = tp; }
        }
    }
    __shared__ float cand[128 * KNN];
    #pragma unroll
    for (int t = 0; t < KNN; ++t) cand[tid * KNN + t] = best[t];
    __syncthreads();
    if (tid == 0) {
        float fin[KNN];
        #pragma unroll
        for (int t = 0; t < KNN; ++t) fin[t] = FLT_BIG;
        for (int i = 0; i < 128 * KNN; ++i) {
            float d = cand[i];
            if (d < fin[KNN - 1]) {
                fin[KNN - 1] = d;
                for (int t = KNN - 1; t > 0; --t)
                    if (fin[t] < fin[t - 1]) { float tp = fin[t]; fin[t] = fin[t - 1]; fin[t - 1] = tp; }
            }
        }
        float s = 0.f;
        for (int t = 0; t < KNN; ++t) s += fin[t] * fin[t];
        float dens = expf(-s / (float)KNN);
        // deterministic tiny tie-break (stands in for jax uniform*1e-6)
        unsigned v = (unsigned)row;
        v ^= v >> 16; v *= 0x7feb352dU; v ^= v >> 15; v *= 0x846ca68bU; v ^= v >> 16;
        density[row] = dens + (float)v * 2.3283064365386963e-10f * 1e-6f;
    }
}

// ---------------- 4. dist_min & score ----------------
__global__ void k_score(const float* __restrict__ dist, const float* __restrict__ density,
                        float* __restrict__ score) {
    int row = blockIdx.x;              // b*N + i
    int b   = row >> 10;
    int tid = threadIdx.x;             // 256
    const float* dr = dist + (size_t)row * NN;
    const float* db = density + (size_t)b * NN;
    float di = db[row & (NN - 1)];
    float mx = -FLT_BIG, cmin = FLT_BIG;
    for (int j = tid; j < NN; j += 256) {
        float d = dr[j];
        mx = fmaxf(mx, d);
        if (db[j] > di) cmin = fminf(cmin, d);
    }
    __shared__ float rmx[256], rmn[256];
    rmx[tid] = mx; rmn[tid] = cmin; __syncthreads();
    for (int off = 128; off > 0; off >>= 1) {
        if (tid < off) {
            rmx[tid] = fmaxf(rmx[tid], rmx[tid + off]);
            rmn[tid] = fminf(rmn[tid], rmn[tid + off]);
        }
        __syncthreads();
    }
    if (tid == 0) score[row] = fminf(rmx[0], rmn[0]) * di;
}

// ---------------- 5. top-256 per batch (iterative argmax, lowest-index ties) ----------------
__global__ void k_topk(const float* __restrict__ score, int* __restrict__ index_down) {
    int b = blockIdx.x, tid = threadIdx.x;   // 256
    __shared__ float sc[NN];
    __shared__ float rv[256];
    __shared__ int   ri[256];
    for (int j = tid; j < NN; j += 256) sc[j] = score[b * NN + j];
    __syncthreads();
    for (int kk = 0; kk < KC; ++kk) {
        float bv = -FLT_BIG; int bi = 0;
        for (int j = tid; j < NN; j += 256) {
            float v = sc[j];
            if (v > bv) { bv = v; bi = j; }
        }
        rv[tid] = bv; ri[tid] = bi; __syncthreads();
        for (int off = 128; off > 0; off >>= 1) {
            if (tid < off) {
                float v2 = rv[tid + off]; int i2 = ri[tid + off];
                if (v2 > rv[tid] || (v2 == rv[tid] && i2 < ri[tid])) { rv[tid] = v2; ri[tid] = i2; }
            }
            __syncthreads();
        }
        if (tid == 0) { index_down[b * KC + kk] = ri[0]; sc[ri[0]] = -FLT_BIG; }
        __syncthreads();
    }
}

// ---------------- 6. assign tokens to nearest center ----------------
__global__ void k_assign(const float* __restrict__ dist, const int* __restrict__ index_down,
                         int* __restrict__ idx_cluster) {
    int gid = blockIdx.x * blockDim.x + threadIdx.x;   // B*N
    int b = gid >> 10, n = gid & (NN - 1);
    const float* db = dist + (size_t)b * NN * NN;
    float best = FLT_BIG; int bk = 0;
    for (int kk = 0; kk < KC; ++kk) {
        int c = index_down[b * KC + kk];
        float d = db[(size_t)c * NN + n];
        if (d < best) { best = d; bk = kk; }
    }
    idx_cluster[gid] = bk;
}

// ---------------- 7. centers get their own cluster id ----------------
__global__ void k_centers(const int* __restrict__ index_down, int* __restrict__ idx_cluster) {
    int gid = blockIdx.x * blockDim.x + threadIdx.x;   // B*KC
    int b = gid >> 8, kk = gid & (KC - 1);
    idx_cluster[b * NN + index_down[gid]] = kk;
}

// ---------------- 8. init outputs / weights ----------------
__global__ void k_init(float* __restrict__ out_x, float* __restrict__ all_weight) {
    int gid = blockIdx.x * blockDim.x + threadIdx.x;
    const int total = BB * KC * CC;
    for (int i = gid; i < total; i += gridDim.x * blockDim.x) out_x[i] = 0.f;
    if (gid < BB * KC) all_weight[gid] = 1e-6f;
}

// ---------------- 9. cluster counts ----------------
__global__ void k_counts(const int* __restrict__ idx_cluster, float* __restrict__ all_weight) {
    int gid = blockIdx.x * blockDim.x + threadIdx.x;   // B*N
    int b = gid >> 10;
    atomicAdd(&all_weight[b * KC + idx_cluster[gid]], 1.0f);
}

// ---------------- 10. weighted scatter into x_merged ----------------
__global__ void k_scatter(const float* __restrict__ x, const int* __restrict__ idx_cluster,
                          const float* __restrict__ all_weight, float* __restrict__ out_x) {
    int row = blockIdx.x;              // b*N + n
    int tid = threadIdx.x;             // 256
    int b = row >> 10;
    int k = idx_cluster[row];
    float w = 1.0f / all_weight[b * KC + k];
    const float* src = x + (size_t)row * CC;
    float* dst = out_x + ((size_t)b * KC + k) * CC;
    for (int c = tid; c < CC; c += 256) atomicAdd(&dst[c], src[c] * w);
}

// ---------------- 11. idx_token_new / agg_weight_new ----------------
__global__ void k_final(const int* __restrict__ idx_token, const float* __restrict__ agg_weight,
                        const int* __restrict__ idx_cluster, const float* __restrict__ all_weight,
                        float* __restrict__ out_idx, float* __restrict__ out_agg) {
    int gid = blockIdx.x * blockDim.x + threadIdx.x;   // B*N
    int b = gid >> 10;
    int t = idx_token[gid];
    int cl = idx_cluster[b * NN + t];
    out_idx[gid] = (float)cl;
    out_agg[gid] = agg_weight[gid] / all_weight[b * KC + cl];
}

extern "C" void kernel_launch(void* const* d_in, const int* in_sizes, int n_in,
                              void* d_out, int out_size, void* d_ws, size_t ws_size,
                              hipStream_t stream) {
    const float* x          = (const float*)d_in[0];
    const int*   idx_token  = (const int*)d_in[1];
    const float* agg_weight = (const float*)d_in[2];

    float* out     = (float*)d_out;
    float* out_x   = out;                                   // [B,KC,CC]
    float* out_idx = out + (size_t)BB * KC * CC;            // [B,N]
    float* out_agg = out_idx + (size_t)BB * NN;             // [B,N]

    char*  ws = (char*)d_ws;
    size_t off = 0;
    float* dist       = (float*)(ws + off); off += (size_t)BB * NN * NN * 4;  // 64 MB
    float* x2         = (float*)(ws + off); off += (size_t)BB * NN * 4;
    float* density    = (float*)(ws + off); off += (size_t)BB * NN * 4;
    float* score      = (float*)(ws + off); off += (size_t)BB * NN * 4;
    int*   index_down = (int*)  (ws + off); off += (size_t)BB * KC * 4;
    int*   idx_clust  = (int*)  (ws + off); off += (size_t)BB * NN * 4;
    float* all_weight = (float*)(ws + off); off += (size_t)BB * KC * 4;

    k_norms  <<<BB * NN, 256, 0, stream>>>(x, x2);
    dim3 gg(NN / 64, NN / 64, BB);
    k_dist   <<<gg, 128, 0, stream>>>(x, x2, dist);
    k_density<<<BB * NN, 128, 0, stream>>>(dist, density);
    k_score  <<<BB * NN, 256, 0, stream>>>(dist, density, score);
    k_topk   <<<BB, 256, 0, stream>>>(score, index_down);
    k_assign <<<(BB * NN) / 256, 256, 0, stream>>>(dist, index_down, idx_clust);
    k_centers<<<(BB * KC) / 256, 256, 0, stream>>>(index_down, idx_clust);
    k_init   <<<1024, 256, 0, stream>>>(out_x, all_weight);
    k_counts <<<(BB * NN) / 256, 256, 0, stream>>>(idx_clust, all_weight);
    k_scatter<<<BB * NN, 256, 0, stream>>>(x, idx_clust, all_weight, out_x);
    k_final  <<<(BB * NN) / 256, 256, 0, stream>>>(idx_token, agg_weight, idx_clust,
                                                   all_weight, out_idx, out_agg);
}